// Net_45612552683672
// MI455X (gfx1250) — compile-verified
//
#include <hip/hip_runtime.h>
#include <math.h>

// ---------------- problem constants (match reference) ----------------
#define NN      4096
#define F_IN    500
#define HID     64
#define HEADS   8
#define NCLS    10
#define NE      131072
#define ET      (NE + NN)        // edges + self loops = 135168
#define HD      (HEADS * HID)    // 512

// ---------------- workspace layout (in floats) ----------------
#define XP_OFF    ((size_t)0)                         // [N, 512]
#define ASRC_OFF  (XP_OFF   + (size_t)NN * HD)        // [N, 8]
#define ADST_OFF  (ASRC_OFF + (size_t)NN * HEADS)     // [N, 8]
#define AMAX_OFF  (ADST_OFF + (size_t)NN * HEADS)     // [N, 8]
#define DEN_OFF   (AMAX_OFF + (size_t)NN * HEADS)     // [N, 8]
#define EX_OFF    (DEN_OFF  + (size_t)NN * HEADS)     // [Et, 8]
#define H_OFF     (EX_OFF   + (size_t)ET * HEADS)     // [N, 512]

typedef float v2f __attribute__((ext_vector_type(2)));
typedef float v8f __attribute__((ext_vector_type(8)));

// ---------------- small utility kernels ----------------
__global__ void fill_f32(float* __restrict__ p, float v, size_t n) {
  size_t i = (size_t)blockIdx.x * blockDim.x + threadIdx.x;
  size_t stride = (size_t)gridDim.x * blockDim.x;
  for (; i < n; i += stride) p[i] = v;
}

__global__ void zero_f32x4(float4* __restrict__ p, size_t n4) {
  size_t i = (size_t)blockIdx.x * blockDim.x + threadIdx.x;
  size_t stride = (size_t)gridDim.x * blockDim.x;
  float4 z = make_float4(0.f, 0.f, 0.f, 0.f);
  for (; i < n4; i += stride) p[i] = z;
}

__device__ __forceinline__ void edge_sd(const int* __restrict__ ei, int e,
                                        int& s, int& d) {
  if (e < NE) { s = ei[e]; d = ei[NE + e]; }
  else        { s = e - NE; d = e - NE; }       // appended self loops
}

__device__ __forceinline__ void atomicMaxF(float* addr, float val) {
  if (val >= 0.0f) atomicMax((int*)addr, __float_as_int(val));
  else             atomicMin((unsigned int*)addr, __float_as_uint(val));
}

// ---------------- 1) xp = x @ lin_w via V_WMMA_F32_16X16X4_F32 ----------------
// One wave (32 lanes) per 16x16 output tile. M=4096 -> 256 tiles, Ncols=512 -> 32 tiles.
__global__ __launch_bounds__(32)
void gemm_xp_wmma(const float* __restrict__ x, const float* __restrict__ lw,
                  float* __restrict__ xp) {
  const int tile = blockIdx.x;        // 0..8191
  const int tn   = tile & 31;
  const int tm   = tile >> 5;
  const int lane = threadIdx.x;
  const int half = lane >> 4;         // lanes 16-31 hold K+2/K+3
  const int l16  = lane & 15;
  const int m0 = tm * 16, n0 = tn * 16;

  const float* arow = x  + (size_t)(m0 + l16) * F_IN;   // A: row m0+l16
  const float* bcol = lw + (n0 + l16);                   // B: col n0+l16

  v8f c = {};
  for (int kk = 0; kk < F_IN; kk += 4) {                 // 500 = 125 * 4
    const int ka = kk + 2 * half;
    v2f a, b;
    a.x = arow[ka];
    a.y = arow[ka + 1];
    b.x = bcol[(size_t)ka * HD];
    b.y = bcol[(size_t)(ka + 1) * HD];
    c = __builtin_amdgcn_wmma_f32_16x16x4_f32(
            /*neg_a=*/false, a, /*neg_b=*/false, b,
            /*c_mod=*/(short)0, c, /*reuse_a=*/false, /*reuse_b=*/false);
  }

  // C/D layout: VGPR i -> M = m0 + half*8 + i, N = n0 + l16
  float* outp = xp + ((size_t)(m0 + half * 8)) * HD + (n0 + l16);
  #pragma unroll
  for (int i = 0; i < 8; ++i) outp[(size_t)i * HD] = c[i];
}

// ---------------- 2) a_src/a_dst per (node, head) ----------------
__global__ __launch_bounds__(256)
void attn_scores(const float* __restrict__ xp, const float* __restrict__ att_src,
                 const float* __restrict__ att_dst,
                 float* __restrict__ a_src, float* __restrict__ a_dst) {
  int t = blockIdx.x * blockDim.x + threadIdx.x;  // N*HEADS
  if (t >= NN * HEADS) return;
  const int n = t >> 3, h = t & 7;
  const float* xr = xp + (size_t)n * HD + h * HID;
  const float* as = att_src + h * HID;
  const float* ad = att_dst + h * HID;
  float s1 = 0.f, s2 = 0.f;
  #pragma unroll 8
  for (int d = 0; d < HID; ++d) { float v = xr[d]; s1 += v * as[d]; s2 += v * ad[d]; }
  a_src[t] = s1;
  a_dst[t] = s2;
}

// ---------------- 3) edge pass 1: segment max of leaky_relu score ----------------
__global__ __launch_bounds__(256)
void edge_pass1(const int* __restrict__ ei, const float* __restrict__ a_src,
                const float* __restrict__ a_dst, float* __restrict__ amax) {
  int t = blockIdx.x * blockDim.x + threadIdx.x;   // Et*HEADS
  if (t >= ET * HEADS) return;
  const int e = t >> 3, h = t & 7;
  int s, d; edge_sd(ei, e, s, d);
  float a = a_src[s * HEADS + h] + a_dst[d * HEADS + h];
  a = a > 0.f ? a : 0.2f * a;                      // leaky_relu, slope 0.2
  atomicMaxF(&amax[d * HEADS + h], a);
}

// ---------------- 4) edge pass 2: exp + segment sum ----------------
__global__ __launch_bounds__(256)
void edge_pass2(const int* __restrict__ ei, const float* __restrict__ a_src,
                const float* __restrict__ a_dst, const float* __restrict__ amax,
                float* __restrict__ ex, float* __restrict__ den) {
  int t = blockIdx.x * blockDim.x + threadIdx.x;
  if (t >= ET * HEADS) return;
  const int e = t >> 3, h = t & 7;
  int s, d; edge_sd(ei, e, s, d);
  float a = a_src[s * HEADS + h] + a_dst[d * HEADS + h];
  a = a > 0.f ? a : 0.2f * a;
  float m = amax[d * HEADS + h];
  if (!isfinite(m)) m = 0.f;                       // matches reference guard
  float v = __expf(a - m);
  ex[t] = v;
  atomicAdd(&den[d * HEADS + h], v);
}

// ---------------- 5) edge pass 3a: normalize alpha + scatter into att1 ----------------
__global__ __launch_bounds__(256)
void edge_pass3a(const int* __restrict__ ei, const float* __restrict__ den,
                 float* __restrict__ ex /* in: exp, out: alpha */,
                 float* __restrict__ att1) {
  int t = blockIdx.x * blockDim.x + threadIdx.x;
  if (t >= ET * HEADS) return;
  const int e = t >> 3, h = t & 7;
  int s, d; edge_sd(ei, e, s, d);
  float al = ex[t] / (den[d * HEADS + h] + 1e-16f);
  ex[t] = al;
  att1[(size_t)s * NN * HEADS + (size_t)d * HEADS + h] = al;
}

// ---------------- 6) edge pass 3b: h[dst] += alpha * xp[src] ----------------
__global__ __launch_bounds__(256)
void edge_pass3b(const int* __restrict__ ei, const float* __restrict__ xp,
                 const float* __restrict__ alpha, float* __restrict__ hbuf) {
  size_t t = (size_t)blockIdx.x * blockDim.x + threadIdx.x;  // Et * 512
  if (t >= (size_t)ET * HD) return;
  const int e = (int)(t >> 9);
  const int r = (int)(t & (HD - 1));                // h*64 + dim
  int s, d; edge_sd(ei, e, s, d);
  float al = alpha[(size_t)e * HEADS + (r >> 6)];
  float v  = xp[(size_t)s * HD + r];
  atomicAdd(&hbuf[(size_t)d * HD + r], v * al);
}

// ---------------- 7) head: elu((h+bias) @ W) -> log_softmax ----------------
// One wave per node; lanes stripe K=512, shuffle-reduce 10 logits.
__global__ __launch_bounds__(256)
void head_kernel(const float* __restrict__ hbuf, const float* __restrict__ bias,
                 const float* __restrict__ W, float* __restrict__ logp) {
  const int lane = threadIdx.x & 31;
  const int node = blockIdx.x * 8 + (threadIdx.x >> 5);
  if (node >= NN) return;
  float acc[NCLS];
  #pragma unroll
  for (int c = 0; c < NCLS; ++c) acc[c] = 0.f;
  const float* hrow = hbuf + (size_t)node * HD;
  for (int k = lane; k < HD; k += 32) {
    const float hv = hrow[k] + bias[k];
    const float* wr = W + (size_t)k * NCLS;
    #pragma unroll
    for (int c = 0; c < NCLS; ++c) acc[c] += hv * wr[c];
  }
  #pragma unroll
  for (int c = 0; c < NCLS; ++c)
    #pragma unroll
    for (int off = 16; off > 0; off >>= 1)
      acc[c] += __shfl_down(acc[c], off, 32);
  if (lane == 0) {
    float lg[NCLS];
    float mx = -INFINITY;
    #pragma unroll
    for (int c = 0; c < NCLS; ++c) {
      float v = acc[c];
      lg[c] = v > 0.f ? v : expm1f(v);              // elu
      mx = fmaxf(mx, lg[c]);
    }
    float se = 0.f;
    #pragma unroll
    for (int c = 0; c < NCLS; ++c) se += __expf(lg[c] - mx);
    const float lse = mx + __logf(se);
    #pragma unroll
    for (int c = 0; c < NCLS; ++c) logp[(size_t)node * NCLS + c] = lg[c] - lse;
  }
}

// ---------------- launcher ----------------
extern "C" void kernel_launch(void* const* d_in, const int* in_sizes, int n_in,
                              void* d_out, int out_size, void* d_ws, size_t ws_size,
                              hipStream_t stream) {
  (void)in_sizes; (void)n_in; (void)out_size; (void)ws_size;
  const float* x       = (const float*)d_in[0];   // [N, F_IN]
  const int*   ei      = (const int*)  d_in[1];   // [2, E]
  const float* lin_w   = (const float*)d_in[2];   // [F_IN, 512]
  const float* att_src = (const float*)d_in[3];   // [8, 64]
  const float* att_dst = (const float*)d_in[4];   // [8, 64]
  const float* bias    = (const float*)d_in[5];   // [512]
  const float* W       = (const float*)d_in[6];   // [512, 10]

  float* out  = (float*)d_out;
  float* logp = out;                               // [N, 10]
  float* att1 = out + (size_t)NN * NCLS;           // [N, N, 8]

  float* ws    = (float*)d_ws;
  float* xp    = ws + XP_OFF;
  float* a_src = ws + ASRC_OFF;
  float* a_dst = ws + ADST_OFF;
  float* amax  = ws + AMAX_OFF;
  float* den   = ws + DEN_OFF;
  float* ex    = ws + EX_OFF;
  float* hbuf  = ws + H_OFF;

  // init: att1 zero (537 MB, the roofline floor), h zero, den zero, amax=-inf
  zero_f32x4<<<4096, 256, 0, stream>>>((float4*)att1, ((size_t)NN * NN * HEADS) / 4);
  fill_f32 <<<2048, 256, 0, stream>>>(hbuf, 0.f, (size_t)NN * HD);
  fill_f32 <<< 128, 256, 0, stream>>>(den,  0.f, (size_t)NN * HEADS);
  fill_f32 <<< 128, 256, 0, stream>>>(amax, -INFINITY, (size_t)NN * HEADS);

  // xp = x @ lin_w  (f32 WMMA, one wave per 16x16 tile)
  gemm_xp_wmma<<<(NN / 16) * (HD / 16), 32, 0, stream>>>(x, lin_w, xp);

  // per-node attention scores
  attn_scores<<<(NN * HEADS) / 256, 256, 0, stream>>>(xp, att_src, att_dst, a_src, a_dst);

  // edge softmax passes
  const int etb = (ET * HEADS) / 256;              // 4224 blocks exactly
  edge_pass1 <<<etb, 256, 0, stream>>>(ei, a_src, a_dst, amax);
  edge_pass2 <<<etb, 256, 0, stream>>>(ei, a_src, a_dst, amax, ex, den);
  edge_pass3a<<<etb, 256, 0, stream>>>(ei, den, ex, att1);

  // message aggregation: 69M L2-resident f32 atomics
  const size_t agg = (size_t)ET * HD;              // 69,206,016 = 270336 * 256
  edge_pass3b<<<(unsigned)(agg / 256), 256, 0, stream>>>(ei, xp, ex, hbuf);

  // classifier head + log_softmax
  head_kernel<<<NN / 8, 256, 0, stream>>>(hbuf, bias, W, logp);
}